// Net_61649960567451
// MI455X (gfx1250) — compile-verified
//
#include <hip/hip_runtime.h>
#include <cstdint>
#include <cstddef>

// ---------------- problem constants (from reference) ----------------
#define N_    65536
#define E_    524288
#define B_    256
#define FEAT_ 64
#define EMB_  128
#define NH_   4
#define NB_   2
#define NC_   4
#define OUTD_ 32
#define SLOPE_ 0.01f
#define NPG_  256   // nodes per graph (N/B), batch is contiguous: i // 256

typedef float v2f __attribute__((ext_vector_type(2)));
typedef float v8f __attribute__((ext_vector_type(8)));

__device__ __forceinline__ float lrelu_(float v) { return v > 0.f ? v : SLOPE_ * v; }

// order-preserving float<->uint encoding for atomic max (sentinel 0 == "-inf / untouched")
__device__ __forceinline__ unsigned fenc(float f) {
  unsigned u = __float_as_uint(f);
  return (u & 0x80000000u) ? ~u : (u | 0x80000000u);
}
__device__ __forceinline__ float fdec(unsigned k) {
  unsigned u = (k & 0x80000000u) ? (k ^ 0x80000000u) : ~k;
  return __uint_as_float(u);
}

// ---------------- generic fill ----------------
__global__ void fill_u32(unsigned* __restrict__ p, unsigned v, size_t count) {
  size_t i = (size_t)blockIdx.x * blockDim.x + threadIdx.x;
  if (i < count) p[i] = v;
}

// ---------------- edge segment-max aggregation (atomic, L2-resident) ----------------
// one thread per (edge, feature); D = 1<<dshift
__global__ void edge_max_agg(const float* __restrict__ X, const int* __restrict__ src,
                             const int* __restrict__ dst, const int* __restrict__ keep,
                             unsigned* __restrict__ aggK, int dshift) {
  size_t idx = (size_t)blockIdx.x * blockDim.x + threadIdx.x;
  int e = (int)(idx >> dshift);
  int d = (int)(idx & (((size_t)1 << dshift) - 1));
  int s = src[e], t = dst[e];
  if (keep && !(keep[s] && keep[t])) return;
  float v = X[((size_t)s << dshift) + d];
  atomicMax(aggK + ((size_t)t << dshift) + d, fenc(v));
}

// in-place decode: key -> float (0 where no in-edge, matching PyG -inf->0 fill)
__global__ void decode_agg(unsigned* __restrict__ p, size_t count) {
  size_t i = (size_t)blockIdx.x * blockDim.x + threadIdx.x;
  if (i >= count) return;
  unsigned k = p[i];
  ((float*)p)[i] = (k == 0u) ? 0.f : fdec(k);
}

// ---------------- fused dual-GEMM with fp32 WMMA (fully unrolled) ----------------
// out[row, colOff+col] = maskrow( lrelu( A1[N,K] @ W1[M,K]^T  (+ A2 @ W2^T)  + bias ) )
// One wave computes a 16x16 C tile using V_WMMA_F32_16X16X4_F32 (exact fp32).
// K, M compile-time -> straight-line WMMA chain, no inner-loop SALU/branches.
template<int K, int M, bool HAS2, bool MASK>
__global__ __launch_bounds__(128)
void gemm_wmma(const float* __restrict__ A1, const float* __restrict__ A2,
               const float* __restrict__ W1, const float* __restrict__ W2,
               const float* __restrict__ bias, const int* __restrict__ keep,
               float* __restrict__ out, int ldOut, int colOff) {
  const int lane = threadIdx.x & 31;
  const int wave = threadIdx.x >> 5;
  const int tile = blockIdx.x * (blockDim.x >> 5) + wave;
  constexpr int colTiles = M >> 4;
  const int rowTile = tile / colTiles;
  const int colTile = tile - rowTile * colTiles;
  const int row0 = rowTile << 4;
  const int col0 = colTile << 4;
  const int r  = lane & 15;     // A row within tile; also B/C column within tile
  const int kh = lane >> 4;     // K-pair selector per ISA fragment layout

  v8f acc = {};
  const float* a1p = A1 + (size_t)(row0 + r) * K + 2 * kh;
  const float* b1p = W1 + (size_t)(col0 + r) * K + 2 * kh;   // B[k,m] = W[m*K + k]
  const float* a2p = HAS2 ? A2 + (size_t)(row0 + r) * K + 2 * kh : nullptr;
  const float* b2p = HAS2 ? W2 + (size_t)(col0 + r) * K + 2 * kh : nullptr;

  #pragma unroll
  for (int k0 = 0; k0 < K; k0 += 4) {
    v2f a = *(const v2f*)(a1p + k0);
    v2f b = *(const v2f*)(b1p + k0);
    acc = __builtin_amdgcn_wmma_f32_16x16x4_f32(false, a, false, b, (short)0, acc, false, false);
    if constexpr (HAS2) {
      v2f a2 = *(const v2f*)(a2p + k0);
      v2f b2 = *(const v2f*)(b2p + k0);
      acc = __builtin_amdgcn_wmma_f32_16x16x4_f32(false, a2, false, b2, (short)0, acc, false, false);
    }
  }

  const float bv = bias[col0 + r];
  #pragma unroll
  for (int vr = 0; vr < 8; ++vr) {
    int row = row0 + vr + 8 * kh;   // C layout: VGPR vr, lane half selects M or M+8
    float v = lrelu_(acc[vr] + bv);
    if constexpr (MASK) { if (!keep[row]) v = 0.f; }
    out[(size_t)row * ldOut + colOff + r] = v;
  }
}

// ---------------- row dot: out[i] = X[i,0:128] . w (+ bias[0]) ----------------
__global__ void dot128(const float* __restrict__ X, const float* __restrict__ w,
                       const float* __restrict__ biasPtr, float* __restrict__ out) {
  int idx = blockIdx.x * blockDim.x + threadIdx.x;
  int node = idx >> 5, lane = idx & 31;
  const float* row = X + (size_t)node * EMB_;
  float s = row[lane] * w[lane] + row[lane + 32] * w[lane + 32]
          + row[lane + 64] * w[lane + 64] + row[lane + 96] * w[lane + 96];
  for (int off = 16; off; off >>= 1) s += __shfl_xor(s, off, 32);
  if (lane == 0) out[node] = s + (biasPtr ? biasPtr[0] : 0.f);
}

// ---------------- block (per-graph) reductions ----------------
__device__ __forceinline__ float blk_max(float v, float* red) {
  int t = threadIdx.x;
  red[t] = v; __syncthreads();
  for (int s = 128; s; s >>= 1) { if (t < s) red[t] = fmaxf(red[t], red[t + s]); __syncthreads(); }
  float r = red[0]; __syncthreads();
  return r;
}
__device__ __forceinline__ float blk_sum(float v, float* red) {
  int t = threadIdx.x;
  red[t] = v; __syncthreads();
  for (int s = 128; s; s >>= 1) { if (t < s) red[t] += red[t + s]; __syncthreads(); }
  float r = red[0]; __syncthreads();
  return r;
}

// TopK pooling: per-graph softmax of scores, keep = sm > min(smax-1e-7, min_score)
__global__ void topk_softmax(const float* __restrict__ score, float* __restrict__ sm,
                             int* __restrict__ keep, float min_score) {
  __shared__ float red[NPG_];
  int i = blockIdx.x * NPG_ + threadIdx.x;
  float z = score[i];
  float m = blk_max(z, red);
  float e = expf(z - m);
  float s = blk_sum(e, red);
  float v = e / (s + 1e-16f);
  float smax = blk_max(v, red);
  float thr = fminf(smax - 1e-7f, min_score);
  sm[i] = v;
  keep[i] = (v > thr) ? 1 : 0;
}

// masked per-graph softmax for global attention
__global__ void att_softmax(const float* __restrict__ gate, const int* __restrict__ keep,
                            float* __restrict__ a) {
  __shared__ float red[NPG_];
  int i = blockIdx.x * NPG_ + threadIdx.x;
  int k = keep[i];
  float z = k ? gate[i] : -__builtin_inff();
  float m = blk_max(z, red);
  if (m == -__builtin_inff()) m = 0.f;
  float e = k ? expf(z - m) : 0.f;
  float s = blk_sum(e, red);
  a[i] = e / (s + 1e-16f);
}

// xp = keep ? cur * sm : 0
__global__ void scale_mask(const float* __restrict__ X, const float* __restrict__ sm,
                           const int* __restrict__ keep, float* __restrict__ xp) {
  size_t idx = (size_t)blockIdx.x * blockDim.x + threadIdx.x;
  int node = (int)(idx >> 7);
  xp[idx] = keep[node] ? X[idx] * sm[node] : 0.f;
}

// ---------------- BatchNorm (training-mode, biased var) ----------------
__global__ void bn_stats(const float* __restrict__ X, float* __restrict__ sum,
                         float* __restrict__ sumsq) {
  int c = threadIdx.x;                       // 128 channels
  size_t base = (size_t)blockIdx.x * 512;    // 512 rows per block
  float s = 0.f, q = 0.f;
  for (int r = 0; r < 512; ++r) {
    float v = X[(base + r) * EMB_ + c];
    s += v; q += v * v;
  }
  atomicAdd(&sum[c], s);
  atomicAdd(&sumsq[c], q);
}

// resid==null: out = bn(X);   resid!=null: out = 0.5*resid + 0.25*bn(X)
__global__ void bn_apply(const float* __restrict__ X, const float* __restrict__ sum,
                         const float* __restrict__ sumsq, const float* __restrict__ g,
                         const float* __restrict__ b, float* __restrict__ out,
                         const float* __restrict__ resid) {
  size_t idx = (size_t)blockIdx.x * blockDim.x + threadIdx.x;
  int c = (int)(idx & (EMB_ - 1));
  const float n = (float)N_;
  float mu = sum[c] / n;
  float var = sumsq[c] / n - mu * mu;
  float y = g[c] * (X[idx] - mu) * rsqrtf(var + 1e-5f) + b[c];
  out[idx] = resid ? 0.5f * resid[idx] + 0.25f * y : y;
}

// ---------------- attention-weighted segment sum: pooled[g,c] = sum_n a*xp ----------------
__global__ void att_pool(const float* __restrict__ a, const float* __restrict__ xp,
                         float* __restrict__ pooled) {
  __shared__ float sa[NPG_];
  int g = blockIdx.x, c = threadIdx.x;           // 128 threads
  sa[c] = a[g * NPG_ + c];
  sa[c + 128] = a[g * NPG_ + c + 128];
  __syncthreads();
  const float* xrow = xp + (size_t)g * NPG_ * EMB_;
  float s = 0.f;
  for (int n = 0; n < NPG_; ++n) s += sa[n] * xrow[(size_t)n * EMB_ + c];
  pooled[g * EMB_ + c] = s;
}

// ---------------- per-graph classifier MLP: 128 -> lrelu -> dot -> scalar ----------------
__global__ void cls_mlp(const float* __restrict__ pooled, const float* __restrict__ fW1,
                        const float* __restrict__ fb1, const float* __restrict__ fW2,
                        const float* __restrict__ fb2, float* __restrict__ logits, int cIdx) {
  __shared__ float pl[EMB_];
  __shared__ float sh[EMB_];
  int g = blockIdx.x, j = threadIdx.x;           // 128 threads
  pl[j] = pooled[g * EMB_ + j];
  __syncthreads();
  float h = fb1[j];
  const float* wrow = fW1 + (size_t)j * EMB_;
  for (int k = 0; k < EMB_; ++k) h += pl[k] * wrow[k];
  sh[j] = lrelu_(h) * fW2[j];
  __syncthreads();
  for (int s = 64; s; s >>= 1) { if (j < s) sh[j] += sh[j + s]; __syncthreads(); }
  if (j == 0) logits[g * NC_ + cIdx] = sh[0] + fb2[0];
}

__global__ void log_softmax4(const float* __restrict__ logits, float* __restrict__ out) {
  int g = threadIdx.x;                            // 256 threads, 1 block
  float z0 = logits[g * 4 + 0], z1 = logits[g * 4 + 1];
  float z2 = logits[g * 4 + 2], z3 = logits[g * 4 + 3];
  float m = fmaxf(fmaxf(z0, z1), fmaxf(z2, z3));
  float l = m + logf(expf(z0 - m) + expf(z1 - m) + expf(z2 - m) + expf(z3 - m));
  out[g * 4 + 0] = z0 - l; out[g * 4 + 1] = z1 - l;
  out[g * 4 + 2] = z2 - l; out[g * 4 + 3] = z3 - l;
}

// ====================================================================
extern "C" void kernel_launch(void* const* d_in, const int* in_sizes, int n_in,
                              void* d_out, int out_size, void* d_ws, size_t ws_size,
                              hipStream_t stream) {
  (void)in_sizes; (void)n_in; (void)out_size; (void)ws_size;
  const float* x       = (const float*)d_in[0];
  const int*   src     = (const int*)d_in[1];
  const int*   dst     = src + E_;
  const float* Wrel0   = (const float*)d_in[3];
  const float* Wroot0  = (const float*)d_in[4];
  const float* b0      = (const float*)d_in[5];
  const float* bn0g    = (const float*)d_in[6];
  const float* bn0b    = (const float*)d_in[7];
  const float* blkPw   = (const float*)d_in[8];    // [NB,NH,EMB]
  const float* blkWrel = (const float*)d_in[9];    // [NB,NH,OUT,EMB]
  const float* blkWroot= (const float*)d_in[10];
  const float* blkB    = (const float*)d_in[11];   // [NB,NH,OUT]
  const float* blkBNg  = (const float*)d_in[12];   // [NB,EMB]
  const float* blkBNb  = (const float*)d_in[13];
  const float* clsPw   = (const float*)d_in[14];   // [NC,EMB]
  const float* gW1     = (const float*)d_in[15];   // [NC,EMB,EMB]
  const float* gb1     = (const float*)d_in[16];
  const float* gW2     = (const float*)d_in[17];
  const float* gb2     = (const float*)d_in[18];
  const float* fW1     = (const float*)d_in[19];
  const float* fb1     = (const float*)d_in[20];
  const float* fW2     = (const float*)d_in[21];
  const float* fb2     = (const float*)d_in[22];

  // workspace layout (all L2-resident, ~135 MB)
  float*    cur   = (float*)d_ws;                          // N*EMB
  float*    xp    = cur + (size_t)N_ * EMB_;               // N*EMB
  unsigned* aggK  = (unsigned*)(xp + (size_t)N_ * EMB_);   // N*EMB (aliased float after decode)
  float*    aggF  = (float*)aggK;
  float*    nb    = (float*)(aggK + (size_t)N_ * EMB_);    // N*EMB (concat heads / gate hidden)
  float*    score = nb + (size_t)N_ * EMB_;                // N
  float*    sm    = score + N_;                            // N
  int*      keep  = (int*)(sm + N_);                       // N
  float*    bnsum = (float*)(keep + N_);                   // EMB
  float*    bnsq  = bnsum + EMB_;                          // EMB
  float*    pooled= bnsq + EMB_;                           // B*EMB
  float*    logits= pooled + (size_t)B_ * EMB_;            // B*NC

  // ---------------- conv0 -> lrelu -> bn0 ----------------
  fill_u32<<<(N_ * FEAT_) / 256, 256, 0, stream>>>(aggK, 0u, (size_t)N_ * FEAT_);
  edge_max_agg<<<(unsigned)(((size_t)E_ * FEAT_) / 256), 256, 0, stream>>>(x, src, dst, nullptr, aggK, 6);
  decode_agg<<<(N_ * FEAT_) / 256, 256, 0, stream>>>(aggK, (size_t)N_ * FEAT_);
  gemm_wmma<FEAT_, EMB_, true, false><<<(N_ / 16) * (EMB_ / 16) / 4, 128, 0, stream>>>(
      aggF, x, Wrel0, Wroot0, b0, nullptr, nb, EMB_, 0);
  fill_u32<<<1, 256, 0, stream>>>((unsigned*)bnsum, 0u, 2 * EMB_);
  bn_stats<<<N_ / 512, 128, 0, stream>>>(nb, bnsum, bnsq);
  bn_apply<<<(N_ * EMB_) / 256, 256, 0, stream>>>(nb, bnsum, bnsq, bn0g, bn0b, cur, nullptr);

  // ---------------- NB blocks x NH heads ----------------
  for (int bi = 0; bi < NB_; ++bi) {
    for (int hi = 0; hi < NH_; ++hi) {
      int hb = bi * NH_ + hi;
      dot128<<<(N_ * 32) / 256, 256, 0, stream>>>(cur, blkPw + (size_t)hb * EMB_, nullptr, score);
      topk_softmax<<<B_, NPG_, 0, stream>>>(score, sm, keep, 0.7f);
      scale_mask<<<(N_ * EMB_) / 256, 256, 0, stream>>>(cur, sm, keep, xp);
      fill_u32<<<(N_ * EMB_) / 256, 256, 0, stream>>>(aggK, 0u, (size_t)N_ * EMB_);
      edge_max_agg<<<(unsigned)(((size_t)E_ * EMB_) / 256), 256, 0, stream>>>(xp, src, dst, keep, aggK, 7);
      decode_agg<<<(N_ * EMB_) / 256, 256, 0, stream>>>(aggK, (size_t)N_ * EMB_);
      gemm_wmma<EMB_, OUTD_, true, true><<<(N_ / 16) * (OUTD_ / 16) / 4, 128, 0, stream>>>(
          aggF, xp,
          blkWrel  + (size_t)hb * OUTD_ * EMB_,
          blkWroot + (size_t)hb * OUTD_ * EMB_,
          blkB     + (size_t)hb * OUTD_,
          keep, nb, EMB_, hi * OUTD_);
    }
    fill_u32<<<1, 256, 0, stream>>>((unsigned*)bnsum, 0u, 2 * EMB_);
    bn_stats<<<N_ / 512, 128, 0, stream>>>(nb, bnsum, bnsq);
    bn_apply<<<(N_ * EMB_) / 256, 256, 0, stream>>>(nb, bnsum, bnsq,
        blkBNg + (size_t)bi * EMB_, blkBNb + (size_t)bi * EMB_, cur, cur);
  }

  // ---------------- NC classification heads ----------------
  for (int c = 0; c < NC_; ++c) {
    dot128<<<(N_ * 32) / 256, 256, 0, stream>>>(cur, clsPw + (size_t)c * EMB_, nullptr, score);
    topk_softmax<<<B_, NPG_, 0, stream>>>(score, sm, keep, 0.8f);
    scale_mask<<<(N_ * EMB_) / 256, 256, 0, stream>>>(cur, sm, keep, xp);
    // gate hidden = lrelu(xp @ gW1^T + gb1)  (128x128 GEMM via WMMA)
    gemm_wmma<EMB_, EMB_, false, false><<<(N_ / 16) * (EMB_ / 16) / 4, 128, 0, stream>>>(
        xp, nullptr, gW1 + (size_t)c * EMB_ * EMB_, nullptr,
        gb1 + (size_t)c * EMB_, nullptr, nb, EMB_, 0);
    // gate = hidden . gW2 + gb2
    dot128<<<(N_ * 32) / 256, 256, 0, stream>>>(nb, gW2 + (size_t)c * EMB_, gb2 + c, score);
    att_softmax<<<B_, NPG_, 0, stream>>>(score, keep, sm);
    att_pool<<<B_, 128, 0, stream>>>(sm, xp, pooled);
    cls_mlp<<<B_, 128, 0, stream>>>(pooled, fW1 + (size_t)c * EMB_ * EMB_,
                                    fb1 + (size_t)c * EMB_, fW2 + (size_t)c * EMB_,
                                    fb2 + c, logits, c);
  }

  log_softmax4<<<1, B_, 0, stream>>>(logits, (float*)d_out);
}